// MLPPredictor_41918880809002
// MI455X (gfx1250) — compile-verified
//
#include <hip/hip_runtime.h>

typedef float v2f __attribute__((ext_vector_type(2)));
typedef float v8f __attribute__((ext_vector_type(8)));

#define D 128
#define WAVES_PER_BLOCK 8

// ---------------------------------------------------------------------------
// Kernel 1: node projection via V_WMMA_F32_16X16X4_F32 (exact fp32 WMMA).
//   S[n][0] = h[n] . W[0, 0:128]    (src proj, class 0)
//   S[n][1] = h[n] . W[1, 0:128]    (src proj, class 1)
//   S[n][2] = h[n] . W[0, 128:256]  (dst proj, class 0)
//   S[n][3] = h[n] . W[1, 128:256]  (dst proj, class 1)
// One wave32 per 16-row tile of h; N padded 4 -> 16 with zero B columns.
// HBM-bound: ~51 MB of h in, 1.6 MB of S out  (~2.3 us at 23.3 TB/s).
// ---------------------------------------------------------------------------
__global__ __launch_bounds__(256) void node_proj_wmma(
    const float* __restrict__ h, const float* __restrict__ W,
    float* __restrict__ S, int n_nodes, int n_tiles)
{
  // B[n][k] staged in LDS: rows 0..3 are the 4 projection vectors, 4..15 zero.
  __shared__ float Bs[16 * D];  // 8 KB
  const int tid = threadIdx.x;
  for (int i = tid; i < 16 * D; i += 256) {
    const int n = i >> 7;         // output column 0..15
    const int k = i & (D - 1);    // k 0..127
    float v = 0.0f;
    if (n < 2)      v = W[n * (2 * D) + k];            // W_src rows
    else if (n < 4) v = W[(n - 2) * (2 * D) + D + k];  // W_dst rows
    Bs[i] = v;
  }
  __syncthreads();

  const int wave = tid >> 5;
  const int lane = tid & 31;
  const int tile = blockIdx.x * WAVES_PER_BLOCK + wave;
  if (tile >= n_tiles) return;   // whole-wave guard: EXEC all-1s for WMMA

  const int m  = lane & 15;   // A row within tile / B,D column
  const int kh = lane >> 4;   // which K-half this lane's VGPRs hold

  // A-matrix 16x4 f32 layout: lane holds row m, K = kh*2 + {0,1}
  int arow_idx = tile * 16 + m;
  if (arow_idx >= n_nodes) arow_idx = n_nodes - 1;  // clamp (VALU cndmask, no EXEC change)
  const float* __restrict__ arow = h + (size_t)arow_idx * D + kh * 2;
  // B-matrix 4x16 f32 layout: lane holds column m, same K split
  const float* brow = &Bs[m * D + kh * 2];

  v8f acc = {};
#pragma unroll 4
  for (int k0 = 0; k0 < D; k0 += 4) {
    v2f a = *(const v2f*)(arow + k0);   // global_load_b64 (clause of 4)
    v2f b = *(const v2f*)(brow + k0);   // ds_load_2addr_b64 pairs
    // 8 args: (neg_a, A, neg_b, B, c_mod, C, reuse_a, reuse_b)
    acc = __builtin_amdgcn_wmma_f32_16x16x4_f32(
        false, a, false, b, (short)0, acc, false, false);
  }

  // D layout (16x16 f32, 8 VGPRs): lane holds column N=m; VGPR v -> row M = kh*8 + v.
  // Only columns 0..3 are real output.
  const bool full_tile = (tile * 16 + 16) <= n_nodes;  // uniform; always true here
  if (m < 4) {
    float* __restrict__ srow = S + (size_t)(tile * 16 + kh * 8) * 4 + m;
    if (full_tile) {
#pragma unroll
      for (int v = 0; v < 8; ++v) srow[v * 4] = acc[v];   // 8 plain global_store_b32
    } else {
      const int row0 = tile * 16 + kh * 8;
#pragma unroll
      for (int v = 0; v < 8; ++v)
        if (row0 + v < n_nodes) srow[v * 4] = acc[v];     // cold fallback
    }
  }
}

// ---------------------------------------------------------------------------
// Kernel 2: per-edge gather + add.  S table (1.6 MB) is L2-resident.
//   out[e*2+c] = S[src[e]][c] + S[dst[e]][2+c] + b[c]
// ---------------------------------------------------------------------------
__global__ __launch_bounds__(256) void edge_score(
    const float* __restrict__ S, const int* __restrict__ src,
    const int* __restrict__ dst, const float* __restrict__ b,
    float* __restrict__ out, int n_edges)
{
  const int e = blockIdx.x * blockDim.x + threadIdx.x;
  if (e >= n_edges) return;
  const int si = src[e];
  const int di = dst[e];
  const float4* __restrict__ S4 = (const float4*)S;
  const float4 ss = S4[si];   // global_load_b128 gather
  const float4 sd = S4[di];   // global_load_b128 gather
  float2 o;
  o.x = ss.x + sd.z + b[0];
  o.y = ss.y + sd.w + b[1];
  *(float2*)(out + (size_t)e * 2) = o;   // coalesced b64 store
}

// ---------------------------------------------------------------------------
extern "C" void kernel_launch(void* const* d_in, const int* in_sizes, int n_in,
                              void* d_out, int out_size, void* d_ws, size_t ws_size,
                              hipStream_t stream) {
  const float* h   = (const float*)d_in[0];  // (N_NODES, 128) f32
  const float* W   = (const float*)d_in[1];  // (2, 256) f32
  const float* b   = (const float*)d_in[2];  // (2,) f32
  const int*   src = (const int*)d_in[3];    // (N_EDGES,) i32
  const int*   dst = (const int*)d_in[4];    // (N_EDGES,) i32
  float* out = (float*)d_out;                // (N_EDGES, 2) f32
  float* S   = (float*)d_ws;                 // n_nodes*4 f32 = 1.6 MB scratch

  const int n_nodes = in_sizes[0] / D;       // 100000
  const int n_edges = in_sizes[3];           // 640000
  const int n_tiles = (n_nodes + 15) / 16;   // 6250

  const int blocks1 = (n_tiles + WAVES_PER_BLOCK - 1) / WAVES_PER_BLOCK;
  node_proj_wmma<<<blocks1, 256, 0, stream>>>(h, W, S, n_nodes, n_tiles);

  const int blocks2 = (n_edges + 255) / 256;
  edge_score<<<blocks2, 256, 0, stream>>>(S, src, dst, b, out, n_edges);
}